// AdditiveAttention_72619307041461
// MI455X (gfx1250) — compile-verified
//
#include <hip/hip_runtime.h>
#include <math.h>

// ---------------------------------------------------------------------------
// Additive attention, MI455X (gfx1250, wave32).
//   q = query @ W_q ; k = key @ W_k              (V_WMMA_F32_16X16X4_F32)
//   scores = sum_h W_v[h]*tanh(q[n,h]+k[m,h])    (v_tanh_f32 hot loop)
//   out    = softmax_m(scores) @ value           (V_WMMA_F32_16X16X4_F32,
//                                                 1/rowsum folded into store)
// ---------------------------------------------------------------------------

#define NB     4      // batches
#define NN     512    // queries per batch
#define MM     512    // keys per batch
#define QS     256    // Q_SIZE
#define KS     256    // K_SIZE
#define HH     128    // H
#define DV     256    // D_V

#define QSTR   132    // padded LDS row stride for q tile   (132 % 64 == 4)
#define SSTR   516    // padded LDS row stride for scores   (516 % 64 == 4)

typedef __attribute__((ext_vector_type(2))) float v2f;
typedef __attribute__((ext_vector_type(8))) float v8f;

__device__ __forceinline__ float fast_tanh(float x) {
#if __has_builtin(__builtin_amdgcn_tanhf)
  return __builtin_amdgcn_tanhf(x);     // v_tanh_f32 (gfx1250 TRANS op)
#else
  return tanhf(x);
#endif
}

__device__ __forceinline__ v8f wmma_f32_16x16x4(v2f a, v2f b, v8f c) {
  // (neg_a, A, neg_b, B, c_mod, C, reuse_a, reuse_b)
  return __builtin_amdgcn_wmma_f32_16x16x4_f32(false, a, false, b,
                                               (short)0, c, false, false);
}

// ---------------------------------------------------------------------------
// Kernel 1: fused projections. One wave computes one 16x16 tile of q or k.
// q tiles: 2048/16 * 128/16 = 1024 ; same for k ; 2048 waves total.
// ---------------------------------------------------------------------------
__global__ __launch_bounds__(128)
void addattn_proj_kernel(const float* __restrict__ query,
                         const float* __restrict__ key,
                         const float* __restrict__ Wq,
                         const float* __restrict__ Wk,
                         float* __restrict__ qproj,
                         float* __restrict__ kproj) {
  const int lane = threadIdx.x & 31;
  const int gw   = blockIdx.x * 4 + (threadIdx.x >> 5);   // 0..2047
  const bool is_k = gw >= 1024;
  const int tile  = gw & 1023;
  const int tm    = tile >> 3;        // 0..127  (rows of 16)
  const int tn    = tile & 7;         // 0..7    (cols of 16)

  const float* __restrict__ A = is_k ? key : query;       // (2048,256)
  const float* __restrict__ W = is_k ? Wk  : Wq;          // (256,128)
  float*       __restrict__ O = is_k ? kproj : qproj;     // (2048,128)

  const int r16   = lane & 15;
  const int khalf = (lane >> 4) << 1;                     // 0 or 2
  const int row   = tm * 16 + r16;
  const int col   = tn * 16 + r16;
  const float* __restrict__ arow = A + (size_t)row * QS;

  v8f acc = {};
#pragma unroll 4
  for (int kb = 0; kb < QS; kb += 4) {
    const int kk = kb + khalf;
    v2f a = *(const v2f*)(arow + kk);                     // A[row][kk..kk+1]
    v2f b;
    b.x = W[kk * HH + col];                               // B[kk  ][col]
    b.y = W[(kk + 1) * HH + col];                         // B[kk+1][col]
    acc = wmma_f32_16x16x4(a, b, acc);
  }

  const int rhi = (lane >> 4) << 3;                       // 0 or 8
#pragma unroll
  for (int i = 0; i < 8; ++i)
    O[(size_t)(tm * 16 + i + rhi) * HH + col] = acc[i];
}

// ---------------------------------------------------------------------------
// Kernel 2: one workgroup (256 thr = 8 waves) per (batch, 16-row query block).
//   phase 1: scores via v_tanh_f32 (k rows broadcast across 16 lanes)
//   phase 2: two-stage row max/sum reduction, store exp() in LDS
//   phase 3: exp @ V with WMMA; rows scaled by 1/rowsum at store
// ---------------------------------------------------------------------------
__global__ __launch_bounds__(256)
void addattn_attend_kernel(const float* __restrict__ qproj,  // (NB*NN, HH)
                           const float* __restrict__ kproj,  // (NB*MM, HH)
                           const float* __restrict__ value,  // (NB, MM, DV)
                           const float* __restrict__ Wv,     // (HH)
                           float* __restrict__ out) {        // (NB, NN, DV)
  __shared__ float q_lds[16 * QSTR];     //  8448 B
  __shared__ float s_lds[16 * SSTR];     // 33024 B  (scores -> exp weights)
  __shared__ float wv_lds[HH];
  __shared__ float red[256];
  __shared__ float rmax[16];
  __shared__ float rsum[16];

  const int tid  = threadIdx.x;
  const int b    = blockIdx.x >> 5;            // 0..3
  const int n0   = (blockIdx.x & 31) << 4;     // query row block base

  // ---- stage q tile + W_v into LDS ----
  if (tid < HH) wv_lds[tid] = Wv[tid];
  for (int i = tid; i < 16 * HH; i += 256) {
    const int r = i >> 7, c = i & (HH - 1);
    q_lds[r * QSTR + c] = qproj[(size_t)(b * NN + n0 + r) * HH + c];
  }
  __syncthreads();

  // ---- phase 1: scores. lanes 0..15 of each 16-group share one m row ----
  const int r  = tid & 15;       // local query row
  const int cg = tid >> 4;       // 0..15 column group
  for (int j = 0; j < 32; ++j) {
    const int m = cg + (j << 4);
    const float4* __restrict__ k4 =
        (const float4*)(kproj + (size_t)(b * MM + m) * HH);
    float s = 0.f;
#pragma unroll
    for (int h4 = 0; h4 < HH / 4; ++h4) {
      const float4 kv = k4[h4];
      const int h = h4 << 2;
      s = fmaf(wv_lds[h + 0], fast_tanh(q_lds[r * QSTR + h + 0] + kv.x), s);
      s = fmaf(wv_lds[h + 1], fast_tanh(q_lds[r * QSTR + h + 1] + kv.y), s);
      s = fmaf(wv_lds[h + 2], fast_tanh(q_lds[r * QSTR + h + 2] + kv.z), s);
      s = fmaf(wv_lds[h + 3], fast_tanh(q_lds[r * QSTR + h + 3] + kv.w), s);
    }
    s_lds[r * SSTR + m] = s;
  }
  __syncthreads();

  // ---- phase 2: softmax (max, exp, sum); weights left unnormalized ----
  float pm = -INFINITY;
  for (int j = 0; j < 32; ++j)
    pm = fmaxf(pm, s_lds[r * SSTR + cg + (j << 4)]);
  red[tid] = pm;                                // tid == cg*16 + r
  __syncthreads();
  if (tid < 16) {
    float mx = red[tid];
    for (int c = 1; c < 16; ++c) mx = fmaxf(mx, red[c * 16 + tid]);
    rmax[tid] = mx;
  }
  __syncthreads();
  const float mx = rmax[r];
  float ps = 0.f;
  for (int j = 0; j < 32; ++j) {
    const int idx = r * SSTR + cg + (j << 4);
    const float e = __expf(s_lds[idx] - mx);
    s_lds[idx] = e;
    ps += e;
  }
  red[tid] = ps;
  __syncthreads();
  if (tid < 16) {
    float sm = 0.f;
    for (int c = 0; c < 16; ++c) sm += red[c * 16 + tid];
    rsum[tid] = sm;
  }
  __syncthreads();

  // ---- phase 3: exp @ V, 8 waves x 2 column tiles, K = 512 in steps of 4 ----
  const int wave  = tid >> 5;
  const int lane  = tid & 31;
  const int r16   = lane & 15;
  const int khalf = (lane >> 4) << 1;           // 0 or 2
  const int rhi   = (lane >> 4) << 3;           // 0 or 8
  const float* __restrict__ Vb = value + (size_t)b * MM * DV;

#pragma unroll
  for (int t = 0; t < 2; ++t) {
    const int ct  = wave + (t << 3);            // column tile 0..15
    const int col = ct * 16 + r16;
    v8f acc = {};
    for (int kb = 0; kb < MM; kb += 4) {
      const int kk = kb + khalf;
      v2f a = *(const v2f*)(s_lds + r16 * SSTR + kk);     // attn rows (LDS)
      v2f bv;
      bv.x = Vb[(size_t)kk * DV + col];                   // V[kk  ][col]
      bv.y = Vb[(size_t)(kk + 1) * DV + col];             // V[kk+1][col]
      acc = wmma_f32_16x16x4(a, bv, acc);
    }
#pragma unroll
    for (int i = 0; i < 8; ++i) {
      const int rr = i + rhi;
      out[(size_t)(b * NN + n0 + rr) * DV + col] = acc[i] / rsum[rr];
    }
  }
}

// ---------------------------------------------------------------------------
extern "C" void kernel_launch(void* const* d_in, const int* in_sizes, int n_in,
                              void* d_out, int out_size, void* d_ws, size_t ws_size,
                              hipStream_t stream) {
  const float* query = (const float*)d_in[0];   // (4,512,256)
  const float* key   = (const float*)d_in[1];   // (4,512,256)
  const float* value = (const float*)d_in[2];   // (4,512,256)
  const float* Wq    = (const float*)d_in[3];   // (256,128)
  const float* Wk    = (const float*)d_in[4];   // (256,128)
  const float* Wv    = (const float*)d_in[5];   // (128,)
  float* out = (float*)d_out;                   // (4,512,256)

  float* qproj = (float*)d_ws;                  // 2048*128 floats = 1 MB
  float* kproj = qproj + (size_t)NB * NN * HH;  // +1 MB (ws >= 2 MB)

  // 2048 tiles (q + k), 4 waves per block of 128 threads
  addattn_proj_kernel<<<512, 128, 0, stream>>>(query, key, Wq, Wk, qproj, kproj);

  // one block per (batch, 16-query-row block): 4*32 = 128 blocks
  addattn_attend_kernel<<<NB * (NN / 16), 256, 0, stream>>>(qproj, kproj, value,
                                                            Wv, out);
}